// Bilinear_46660524703902
// MI455X (gfx1250) — compile-verified
//
#include <hip/hip_runtime.h>

typedef __attribute__((ext_vector_type(2))) float v2f;
typedef __attribute__((ext_vector_type(8))) float v8f;

#define LN 128                     // LEFT (= K of GEMM)
#define ON 128                     // OUT  (= N of GEMM)
#define NROWS (128 * 4096)         // 524288 flattened rows
#define ROWBLK 128                 // rows per workgroup per iteration (8 waves x 16)
#define NRB (NROWS / ROWBLK)       // 4096 row blocks
#define GRID_MAIN 1024
#define GRID_COLSUM 1024
#define ROWS_PER_CS (NROWS / GRID_COLSUM)  // 512
#define LDS_STRIDE 132             // 128 + 4 pad: bank step 4 -> conflict-free b64 loads

// ---------------------------------------------------------------------------
// Pass 1a: per-block partial column sums of x2  ->  partial[1024][128]
// ---------------------------------------------------------------------------
__global__ __launch_bounds__(256) void colsum_partial(
    const float* __restrict__ x2, float* __restrict__ partial) {
  __shared__ float red[256];
  const int c   = threadIdx.x & 127;
  const int sub = threadIdx.x >> 7;  // 0 or 1
  const size_t r0 = (size_t)blockIdx.x * ROWS_PER_CS + sub;
  float s = 0.0f;
  for (int i = 0; i < ROWS_PER_CS; i += 2)
    s += x2[(r0 + i) * LN + c];
  red[threadIdx.x] = s;
  __syncthreads();
  if (sub == 0)
    partial[blockIdx.x * LN + c] = red[c] + red[LN + c];
}

// ---------------------------------------------------------------------------
// Pass 1b: reduce partials -> s[128]   (single block, fixed order: deterministic)
// ---------------------------------------------------------------------------
__global__ __launch_bounds__(1024) void colsum_final(
    const float* __restrict__ partial, float* __restrict__ s) {
  __shared__ float red[1024];
  const int c   = threadIdx.x & 127;
  const int sub = threadIdx.x >> 7;  // 0..7
  float a = 0.0f;
  for (int b = sub; b < GRID_COLSUM; b += 8)
    a += partial[b * LN + c];
  red[threadIdx.x] = a;
  __syncthreads();
  if (sub == 0) {
    float t = 0.0f;
    for (int j = 0; j < 8; ++j) t += red[j * LN + c];
    s[c] = t;
  }
}

// ---------------------------------------------------------------------------
// Pass 2: build transposed weight operands
//   A_t[o][l]  = W_l[l][o] + sum_r U[o][l][r] * s[r]
//   Wr_t[o][r] = W_r[r][o]
// ---------------------------------------------------------------------------
__global__ __launch_bounds__(128) void prep_weights(
    const float* __restrict__ U, const float* __restrict__ Wl,
    const float* __restrict__ Wr, const float* __restrict__ s,
    float* __restrict__ At, float* __restrict__ Wrt) {
  __shared__ float ls[128];
  const int o = blockIdx.x;
  const int l = threadIdx.x;
  ls[l] = s[l];
  __syncthreads();
  const float* u = U + ((size_t)o * LN + l) * 128;  // U[o][l][*], contiguous in r
  float acc = Wl[l * ON + o];
  for (int r = 0; r < 128; r += 4)
    acc += u[r] * ls[r] + u[r + 1] * ls[r + 1] + u[r + 2] * ls[r + 2] +
           u[r + 3] * ls[r + 3];
  At[o * LN + l]  = acc;
  Wrt[o * LN + l] = Wr[l * ON + o];
}

// ---------------------------------------------------------------------------
// Pass 3: out[n][o] = x1[n] @ A + x2[n] @ W_r + N*bias[o]
// 8 waves/block; each wave owns a 16x128 strip (8 WMMA 16x16 column tiles).
// A-fragment layout (f32 16x16x4): lane = 16*half + mr,
//   VGPR0 = A[mr][k+2*half], VGPR1 = A[mr][k+2*half+1]
// B-fragment: b.x = B[k+2*half][n0+mr], b.y = B[k+2*half+1][n0+mr]
//   -> with B stored transposed in LDS these are one contiguous b64 load.
// D/C layout: acc[n][v] = D[v + 8*half][n*16 + mr].
// ---------------------------------------------------------------------------
__global__ __launch_bounds__(256) void bilinear_main(
    const float* __restrict__ x1, const float* __restrict__ x2,
    const float* __restrict__ At, const float* __restrict__ Wrt,
    const float* __restrict__ bias, float* __restrict__ out) {
  __shared__ float lA[ON * LDS_STRIDE];  // A_t  (66 KB)
  __shared__ float lW[ON * LDS_STRIDE];  // Wr_t (66 KB)

  const int tid = threadIdx.x;
  for (int idx = tid; idx < ON * LN; idx += 256) {
    const int o = idx >> 7, k = idx & 127;
    lA[o * LDS_STRIDE + k] = At[idx];
    lW[o * LDS_STRIDE + k] = Wrt[idx];
  }
  __syncthreads();

  const int wave = tid >> 5;
  const int lane = tid & 31;
  const int hh   = lane >> 4;   // half-wave select
  const int mr   = lane & 15;

  // bias column values for this lane (bias summed N times in the reference)
  float nb[8];
#pragma unroll
  for (int n = 0; n < 8; ++n) nb[n] = 524288.0f * bias[n * 16 + mr];

  for (int rb = blockIdx.x; rb < NRB; rb += GRID_MAIN) {
    const int rowbase = rb * ROWBLK + wave * 16;

    v8f acc[8];
#pragma unroll
    for (int n = 0; n < 8; ++n)
#pragma unroll
      for (int v = 0; v < 8; ++v) acc[n][v] = nb[n];

    // ---- GEMM 1: x1 @ A ----
    {
      const float* p = x1 + (size_t)(rowbase + mr) * LN + 2 * hh;
      for (int k = 0; k < LN; k += 4) {
        const v2f a = *(const v2f*)(p + k);
#pragma unroll
        for (int n = 0; n < 8; ++n) {
          const v2f b =
              *(const v2f*)&lA[(n * 16 + mr) * LDS_STRIDE + k + 2 * hh];
          acc[n] = __builtin_amdgcn_wmma_f32_16x16x4_f32(
              false, a, false, b, (short)0, acc[n], false, false);
        }
      }
    }
    // ---- GEMM 2: x2 @ W_r ----
    {
      const float* p = x2 + (size_t)(rowbase + mr) * LN + 2 * hh;
      for (int k = 0; k < LN; k += 4) {
        const v2f a = *(const v2f*)(p + k);
#pragma unroll
        for (int n = 0; n < 8; ++n) {
          const v2f b =
              *(const v2f*)&lW[(n * 16 + mr) * LDS_STRIDE + k + 2 * hh];
          acc[n] = __builtin_amdgcn_wmma_f32_16x16x4_f32(
              false, a, false, b, (short)0, acc[n], false, false);
        }
      }
    }

    // ---- store 16x128 strip (coalesced across lanes per row) ----
#pragma unroll
    for (int v = 0; v < 8; ++v) {
      float* po = out + (size_t)(rowbase + v + 8 * hh) * ON + mr;
#pragma unroll
      for (int n = 0; n < 8; ++n) po[n * 16] = acc[n][v];
    }
  }
}

// ---------------------------------------------------------------------------
extern "C" void kernel_launch(void* const* d_in, const int* in_sizes, int n_in,
                              void* d_out, int out_size, void* d_ws,
                              size_t ws_size, hipStream_t stream) {
  const float* x1   = (const float*)d_in[0];  // input_left  [128,4096,128]
  const float* x2   = (const float*)d_in[1];  // input_right [128,4096,128]
  const float* U    = (const float*)d_in[2];  // [128,128,128]
  const float* Wl   = (const float*)d_in[3];  // [128,128]
  const float* Wr   = (const float*)d_in[4];  // [128,128]
  const float* bias = (const float*)d_in[5];  // [128]
  float* out = (float*)d_out;

  float* ws      = (float*)d_ws;
  float* partial = ws;                 // 1024*128 = 131072 floats
  float* s       = ws + 131072;        // 128
  float* At      = ws + 131200;        // 16384
  float* Wrt     = ws + 147584;        // 16384   (total < 660 KB)

  colsum_partial<<<GRID_COLSUM, 256, 0, stream>>>(x2, partial);
  colsum_final<<<1, 1024, 0, stream>>>(partial, s);
  prep_weights<<<128, 128, 0, stream>>>(U, Wl, Wr, s, At, Wrt);
  bilinear_main<<<GRID_MAIN, 256, 0, stream>>>(x1, x2, At, Wrt, bias, out);
}